// InterleaveMLP_15255723835508
// MI455X (gfx1250) — compile-verified
//
#include <hip/hip_runtime.h>

typedef __attribute__((ext_vector_type(2))) float v2f;
typedef __attribute__((ext_vector_type(4))) float v4f;
typedef __attribute__((ext_vector_type(8))) float v8f;

#define CDIM 128

// ---------------------------------------------------------------- zero (128-bit)
__global__ void zero4_kernel(v4f* __restrict__ p, long n4) {
    long i = (long)blockIdx.x * blockDim.x + threadIdx.x;
    if (i < n4) p[i] = (v4f)(0.0f);
}

// ---------------------------------------------------------------- scatter-add (segment sum + counts)
__global__ void scatter_add_kernel(const float* __restrict__ x, const int* __restrict__ cl,
                                   float* s, float* cnt, long nrows) {
    long i = (long)blockIdx.x * blockDim.x + threadIdx.x;
    long total = nrows * CDIM;
    if (i >= total) return;
    long row = i >> 7;           // /128
    int  col = (int)(i & 127);
    int  c   = cl[row];
    float v  = __builtin_nontemporal_load(x + i);   // x streamed exactly once here
    __hip_atomic_fetch_add(&s[(long)c * CDIM + col], v,
                           __ATOMIC_RELAXED, __HIP_MEMORY_SCOPE_AGENT);
    if (col == 0)
        __hip_atomic_fetch_add(&cnt[c], 1.0f,
                               __ATOMIC_RELAXED, __HIP_MEMORY_SCOPE_AGENT);
}

// ---------------------------------------------------------------- WMMA f32 16x16x4 tile core
// One wave: 16 rows x 128 cols, K=128. Software-pipelined: issue the 9 loads for
// step k+4 before consuming step k, so loads stay in flight behind the WMMAs.
// Arow already includes lane offset (+lrow*C + 2*half); Wb = W + lrow*C + 2*half.
template <bool SCALE>
__device__ __forceinline__ void gemm_tile_core(const float* Arow, const float* Wb,
                                               float inv, v8f acc[8]) {
    v2f a_cur = *(const v2f*)(Arow);
    v2f b_cur[8];
#pragma unroll
    for (int j = 0; j < 8; ++j)
        b_cur[j] = *(const v2f*)(Wb + j * 16 * CDIM);

#pragma unroll
    for (int k = 0; k < CDIM; k += 4) {
        v2f a_nxt = {};
        v2f b_nxt[8] = {};
        if (k + 4 < CDIM) {
            a_nxt = *(const v2f*)(Arow + k + 4);
#pragma unroll
            for (int j = 0; j < 8; ++j)
                b_nxt[j] = *(const v2f*)(Wb + j * 16 * CDIM + k + 4);
        }
        v2f a = a_cur;
        if (SCALE) a *= inv;
#pragma unroll
        for (int j = 0; j < 8; ++j)
            acc[j] = __builtin_amdgcn_wmma_f32_16x16x4_f32(
                false, a, false, b_cur[j], (short)0, acc[j], false, false);
        a_cur = a_nxt;
#pragma unroll
        for (int j = 0; j < 8; ++j) b_cur[j] = b_nxt[j];
    }
}

// ---------------------------------------------------------------- T = X * W^T
__global__ void __launch_bounds__(256) gemm_t_kernel(const float* __restrict__ X,
                                                     const float* __restrict__ W,
                                                     float* __restrict__ T, int nrows) {
    const int lane = threadIdx.x & 31;
    const int wave = threadIdx.x >> 5;
    const int lrow = lane & 15;
    const int half = lane >> 4;
    const long i0 = ((long)blockIdx.x * (blockDim.x >> 5) + wave) * 16;
    if (i0 + 16 > nrows) return;                 // wave-uniform: EXEC stays all-1s

    v8f acc[8] = {};
    gemm_tile_core<false>(X + (i0 + lrow) * CDIM + 2 * half,
                          W + lrow * CDIM + 2 * half, 1.0f, acc);
#pragma unroll
    for (int r = 0; r < 8; ++r) {
        long row = i0 + 8 * half + r;
#pragma unroll
        for (int j = 0; j < 8; ++j)
            T[row * CDIM + j * 16 + lrow] = acc[j][r];
    }
}

// ---------------------------------------------------------------- Y = Xres + (S/cnt) * W^T [+ Tg[cl]]
// S may alias Y: each wave fully reads its 16 rows of S (whole K loop) before writing them.
__global__ void __launch_bounds__(256) gemm_mean_kernel(const float* S, const float* __restrict__ CNT,
                                                        const float* __restrict__ W,
                                                        const float* __restrict__ Xres,
                                                        const float* __restrict__ Tg,
                                                        const int* __restrict__ cl,
                                                        float* Y, int nrows) {
    const int lane = threadIdx.x & 31;
    const int wave = threadIdx.x >> 5;
    const int lrow = lane & 15;
    const int half = lane >> 4;
    const long i0 = ((long)blockIdx.x * (blockDim.x >> 5) + wave) * 16;
    if (i0 + 16 > nrows) return;

    const float inv = 1.0f / fmaxf(CNT[i0 + lrow], 1.0f);   // per A-load row
    v8f acc[8] = {};
    gemm_tile_core<true>(S + (i0 + lrow) * CDIM + 2 * half,
                         W + lrow * CDIM + 2 * half, inv, acc);
#pragma unroll
    for (int r = 0; r < 8; ++r) {
        long row = i0 + 8 * half + r;
        long g = cl ? (long)cl[row] : 0;
#pragma unroll
        for (int j = 0; j < 8; ++j) {
            int col = j * 16 + lrow;
            float v = Xres[row * CDIM + col] + acc[j][r];
            if (Tg) v += Tg[g * CDIM + col];
            Y[row * CDIM + col] = v;
        }
    }
}

// ---------------------------------------------------------------- y0 = x0 + t1[cluster1]
// x0 / y0 stream ~1 GB: non-temporal so the 51 MB t1 gather table stays L2-resident.
__global__ void y0_kernel(const float* __restrict__ x0, const float* __restrict__ t1,
                          const int* __restrict__ cl, float* __restrict__ y0, long n0) {
    long i = (long)blockIdx.x * blockDim.x + threadIdx.x;   // v4f index
    long total = n0 * (CDIM / 4);
    if (i >= total) return;
    long row = i >> 5;            // 32 v4f per row
    int  q   = (int)(i & 31);
    long c   = cl[row];
    v4f a = __builtin_nontemporal_load((const v4f*)x0 + i);
    v4f b = ((const v4f*)(t1 + c * CDIM))[q];
    __builtin_nontemporal_store(a + b, (v4f*)y0 + i);
}

// ----------------------------------------------------------------
extern "C" void kernel_launch(void* const* d_in, const int* in_sizes, int n_in,
                              void* d_out, int out_size, void* d_ws, size_t ws_size,
                              hipStream_t stream) {
    const float* x0       = (const float*)d_in[0];
    const float* x1       = (const float*)d_in[1];
    const float* x2       = (const float*)d_in[2];
    const int*   cluster1 = (const int*)d_in[3];
    const int*   cluster2 = (const int*)d_in[4];
    const float* Wf2c0    = (const float*)d_in[5];
    const float* Wf2c1    = (const float*)d_in[6];
    const float* Wc2f0    = (const float*)d_in[7];
    const float* Wc2f1    = (const float*)d_in[8];

    const long N0 = in_sizes[0] / CDIM;
    const long N1 = in_sizes[1] / CDIM;
    const long N2 = in_sizes[2] / CDIM;

    float* y0 = (float*)d_out;
    float* y1 = y0 + N0 * CDIM;          // doubles as s1 accumulator
    float* y2 = y1 + N1 * CDIM;          // doubles as s2 accumulator

    float* t1 = (float*)d_ws;            // N1*C : x1 @ Wc2f0^T
    float* t2 = t1 + N1 * CDIM;          // N2*C : x2 @ Wc2f1^T
    float* c1 = t2 + N2 * CDIM;          // N1 counts
    float* c2 = c1 + N1;                 // N2 counts

    const int TB = 256;

    // 1) zero segment accumulators (in d_out) and counts (in ws)
    {
        long n4 = (N1 + N2) * CDIM / 4;
        zero4_kernel<<<(n4 + TB - 1) / TB, TB, 0, stream>>>((v4f*)y1, n4);
        long c4 = (N1 + N2 + 3) / 4;
        zero4_kernel<<<(c4 + TB - 1) / TB, TB, 0, stream>>>((v4f*)c1, c4);
    }

    // 2) coarse->fine projections: t1 = x1 @ Wc2f0^T, t2 = x2 @ Wc2f1^T
    {
        long tiles1 = N1 / 16, tiles2 = N2 / 16;
        gemm_t_kernel<<<(tiles1 + 7) / 8, TB, 0, stream>>>(x1, Wc2f0, t1, (int)N1);
        gemm_t_kernel<<<(tiles2 + 7) / 8, TB, 0, stream>>>(x2, Wc2f1, t2, (int)N2);
    }

    // 3) segment sums: s1 += x0 per cluster1, s2 += x1 per cluster2
    {
        long e1 = N0 * CDIM, e2 = N1 * CDIM;
        scatter_add_kernel<<<(e1 + TB - 1) / TB, TB, 0, stream>>>(x0, cluster1, y1, c1, N0);
        scatter_add_kernel<<<(e2 + TB - 1) / TB, TB, 0, stream>>>(x1, cluster2, y2, c2, N1);
    }

    // 4) y0 = x0 + t1[cluster1]
    {
        long q = N0 * (CDIM / 4);
        y0_kernel<<<(q + TB - 1) / TB, TB, 0, stream>>>(x0, t1, cluster1, y0, N0);
    }

    // 5) y1 = x1 + (s1/cnt1) @ Wf2c0^T + t2[cluster2]   (in place over s1)
    {
        long tiles1 = N1 / 16;
        gemm_mean_kernel<<<(tiles1 + 7) / 8, TB, 0, stream>>>(
            y1, c1, Wf2c0, x1, t2, cluster2, y1, (int)N1);
    }

    // 6) y2 = x2 + (s2/cnt2) @ Wf2c1^T                  (in place over s2)
    {
        long tiles2 = N2 / 16;
        gemm_mean_kernel<<<(tiles2 + 7) / 8, TB, 0, stream>>>(
            y2, c2, Wf2c1, x2, nullptr, nullptr, y2, (int)N2);
    }
}